// BANLayer_81999515615424
// MI455X (gfx1250) — compile-verified
//
#include <hip/hip_runtime.h>
#include <math.h>

typedef float v2f __attribute__((ext_vector_type(2)));
typedef float v4f __attribute__((ext_vector_type(4)));
typedef float v8f __attribute__((ext_vector_type(8)));

#define BSZ     64
#define H_DIM_C 512
#define HK      1536
#define V_DIM   512
#define Q_DIM   768
#define OUT_N   1024
#define BN_EPS  1e-5f

// ---------------------------------------------------------------------------
// ws scratch layout (float units):
//   [0]   sumsq(v_W)
//   [1]   sumsq(q_W)
//   [16]  rowsum scaled: s[b] = sum_j v_[b][j] / H_DIM      (64 floats)
//   [128]              v_  buffer  (64*1536)
//   [128+98304]        q_  buffer  (64*1536)
//   [128+2*98304]      logits buffer (64*1536)
// ---------------------------------------------------------------------------
#define WS_SUMSQ   0
#define WS_ROWSUM  16
#define WS_VBUF    128
#define WS_QBUF    (128 + BSZ*HK)
#define WS_LOGITS  (128 + 2*BSZ*HK)

__global__ void zero_kernel(float* w, int n) {
    int i = blockIdx.x * blockDim.x + threadIdx.x;
    if (i < n) w[i] = 0.0f;
}

// Frobenius-norm^2 reduction: grid-stride + block reduce + one f32 atomic.
__global__ void sumsq_kernel(const float* __restrict__ x, int n,
                             float* __restrict__ out) {
    __shared__ float red[256];
    float acc = 0.0f;
    for (int i = blockIdx.x * blockDim.x + threadIdx.x; i < n;
         i += gridDim.x * blockDim.x) {
        float v = x[i];
        acc += v * v;
    }
    red[threadIdx.x] = acc;
    __syncthreads();
    for (int s = blockDim.x >> 1; s > 0; s >>= 1) {
        if (threadIdx.x < (unsigned)s) red[threadIdx.x] += red[threadIdx.x + s];
        __syncthreads();
    }
    if (threadIdx.x == 0) atomicAdd(out, red[0]);
}

// ---------------------------------------------------------------------------
// C[m,n] = act( scale * sum_k A[m,k]*W[n,k] + bias[n] ),  scale = g/||W||_F
// NT GEMM via V_WMMA_F32_16X16X4_F32. One wave -> one 16x16 C tile.
// A fragment (16x4 f32): lane = m + 16*h holds A[m][k0+2h], A[m][k0+2h+1]
// B fragment (4x16 f32): lane = n + 16*h holds W[n][k0+2h], W[n][k0+2h+1]
// C/D (16x16 f32, 8 VGPRs): vgpr r, lane l -> M = r + 8*(l>>4), N = l&15
// ---------------------------------------------------------------------------
__global__ void wn_gemm_wmma(const float* __restrict__ A, int lda,
                             const float* __restrict__ W, int ldw,
                             const float* __restrict__ bias,
                             const float* __restrict__ sumsq,  // null => scale 1
                             const float* __restrict__ gvec,
                             float* __restrict__ C, int ldc,
                             int Kdim, int relu) {
    const int lane = threadIdx.x;          // 0..31, wave32
    const int half = lane >> 4;            // 0/1
    const int lm   = lane & 15;
    const int m0   = threadIdx.y * 16;     // 4 waves cover M = 64
    const int n0   = blockIdx.x * 16;

    const float* aptr = A + (size_t)(m0 + lm) * lda + 2 * half;
    const float* bptr = W + (size_t)(n0 + lm) * ldw + 2 * half;

    v8f acc = {};
    for (int k0 = 0; k0 < Kdim; k0 += 4) {
        v2f a = *(const v2f*)(aptr + k0);
        v2f b = *(const v2f*)(bptr + k0);
        acc = __builtin_amdgcn_wmma_f32_16x16x4_f32(
            /*neg_a=*/false, a, /*neg_b=*/false, b,
            /*c_mod=*/(short)0, acc, /*reuse_a=*/false, /*reuse_b=*/false);
    }

    float scale = 1.0f;
    if (sumsq) scale = gvec[0] * rsqrtf(sumsq[0]);
    const float bn = bias[n0 + lm];

    float* ccol = C + n0 + lm;
    #pragma unroll
    for (int r = 0; r < 8; ++r) {
        int m = m0 + r + 8 * half;
        float v = fmaf(scale, acc[r], bn);
        if (relu) v = fmaxf(v, 0.0f);
        ccol[(size_t)m * ldc] = v;
    }
}

// s[b] = (sum_j v_[b][j]) / H_DIM
__global__ void rowsum_kernel(const float* __restrict__ vbuf,
                              float* __restrict__ s) {
    __shared__ float red[256];
    const int b = blockIdx.x;
    float acc = 0.0f;
    for (int j = threadIdx.x; j < HK; j += blockDim.x)
        acc += vbuf[(size_t)b * HK + j];
    red[threadIdx.x] = acc;
    __syncthreads();
    for (int st = blockDim.x >> 1; st > 0; st >>= 1) {
        if (threadIdx.x < (unsigned)st) red[threadIdx.x] += red[threadIdx.x + st];
        __syncthreads();
    }
    if (threadIdx.x == 0) s[b] = red[0] * (1.0f / (float)H_DIM_C);
}

// logits = BN( s[b] * q_[b][j] ) with running stats (inference)
__global__ void logits_bn_kernel(const float* __restrict__ qbuf,
                                 const float* __restrict__ s,
                                 const float* __restrict__ mean,
                                 const float* __restrict__ var,
                                 const float* __restrict__ gamma,
                                 const float* __restrict__ beta,
                                 float* __restrict__ logits) {
    int i = blockIdx.x * blockDim.x + threadIdx.x;
    if (i >= BSZ * HK) return;
    int b = i / HK;
    int j = i - b * HK;
    float x = s[b] * qbuf[i];
    logits[i] = fmaf((x - mean[j]) * rsqrtf(var[j] + BN_EPS), gamma[j], beta[j]);
}

// att[b][i][j] = v_[b][i] * q_[b][j]  — 604 MB streaming store, NT hints.
__global__ void outer_nt_kernel(const float* __restrict__ vbuf,
                                const float* __restrict__ qbuf,
                                float* __restrict__ att) {
    const int row = blockIdx.x;            // 0 .. BSZ*HK-1  (b,i) pair
    const int b   = row / HK;
    const float vi = vbuf[row];
    const v4f* qrow = (const v4f*)(qbuf + (size_t)b * HK);
    v4f* arow = (v4f*)(att + (size_t)row * HK);
    for (int j4 = threadIdx.x; j4 < HK / 4; j4 += blockDim.x) {
        v4f q4 = qrow[j4];
        v4f o  = vi * q4;
        __builtin_nontemporal_store(o, arow + j4);
    }
}

extern "C" void kernel_launch(void* const* d_in, const int* in_sizes, int n_in,
                              void* d_out, int out_size, void* d_ws, size_t ws_size,
                              hipStream_t stream) {
    (void)in_sizes; (void)n_in; (void)out_size; (void)ws_size;

    const float* v        = (const float*)d_in[0];
    const float* q        = (const float*)d_in[1];
    const float* v_W      = (const float*)d_in[2];
    const float* v_b      = (const float*)d_in[3];
    const float* v_g      = (const float*)d_in[4];
    const float* q_W      = (const float*)d_in[5];
    const float* q_b      = (const float*)d_in[6];
    const float* q_g      = (const float*)d_in[7];
    const float* bn_gamma = (const float*)d_in[8];
    const float* bn_beta  = (const float*)d_in[9];
    const float* bn_mean  = (const float*)d_in[10];
    const float* bn_var   = (const float*)d_in[11];
    const float* fc_W     = (const float*)d_in[12];
    const float* fc_b     = (const float*)d_in[13];

    float* ws     = (float*)d_ws;
    float* sumsq  = ws + WS_SUMSQ;   // [0]=v_W, [1]=q_W
    float* rowsum = ws + WS_ROWSUM;
    float* vbuf   = ws + WS_VBUF;
    float* qbuf   = ws + WS_QBUF;
    float* logits = ws + WS_LOGITS;

    float* out = (float*)d_out;                   // [64,1024]
    float* att = out + (size_t)BSZ * OUT_N;       // [64,1536,1536]

    // 0) zero the atomic accumulators
    zero_kernel<<<1, 32, 0, stream>>>(sumsq, 2);

    // 1) Frobenius norms for weight_norm
    sumsq_kernel<<<256, 256, 0, stream>>>(v_W, HK * V_DIM, &sumsq[0]);
    sumsq_kernel<<<256, 256, 0, stream>>>(q_W, HK * Q_DIM, &sumsq[1]);

    // 2) v_ = relu(scale_v * v @ v_W^T + v_b)   [64,1536]
    wn_gemm_wmma<<<dim3(HK / 16), dim3(32, 4), 0, stream>>>(
        v, V_DIM, v_W, V_DIM, v_b, &sumsq[0], v_g, vbuf, HK, V_DIM, 1);

    // 3) q_ = relu(scale_q * q @ q_W^T + q_b)   [64,1536]
    wn_gemm_wmma<<<dim3(HK / 16), dim3(32, 4), 0, stream>>>(
        q, Q_DIM, q_W, Q_DIM, q_b, &sumsq[1], q_g, qbuf, HK, Q_DIM, 1);

    // 4) s[b] = sum_i v_[b][i] / H_DIM
    rowsum_kernel<<<BSZ, 256, 0, stream>>>(vbuf, rowsum);

    // 5) logits = BN(s[b] * q_)                 [64,1536]
    logits_bn_kernel<<<(BSZ * HK + 255) / 256, 256, 0, stream>>>(
        qbuf, rowsum, bn_mean, bn_var, bn_gamma, bn_beta, logits);

    // 6) out = logits @ fc_W^T + fc_b           [64,1024]
    wn_gemm_wmma<<<dim3(OUT_N / 16), dim3(32, 4), 0, stream>>>(
        logits, HK, fc_W, HK, fc_b, nullptr, nullptr, out, OUT_N, HK, 0);

    // 7) att_maps = outer(v_, q_) — the 604 MB bandwidth-bound stream
    outer_nt_kernel<<<BSZ * HK, 128, 0, stream>>>(vbuf, qbuf, att);
}